// EfficientMoE_64759516889585
// MI455X (gfx1250) — compile-verified
//
#include <hip/hip_runtime.h>
#include <hip/hip_bf16.h>
#include <math.h>

// ---------------------------------------------------------------------------
// EfficientMoE for MI455X (gfx1250), fp32 end-to-end.
//   x[8192,2048], w_router[2048,8], w_fc[8,2048,2048], w_proj[8,2048,2048]
// Pipeline: init -> router(top2+softmax,lists) -> scan -> gemm1(relu^2) ->
//           gemm2(prob-scaled scatter-add).
// GEMMs: v_wmma_f32_16x16x4_f32, 128x128 block tiles, double-buffered LDS
// staging (b128 global loads -> b128 ds stores, conflict-free padded strides).
// ---------------------------------------------------------------------------

#define NTOK   8192
#define CDIM   2048
#define DDIM   2048
#define NEXP   8
#define SEGCAP NTOK
#define BLKTOK 128                  // tokens per block
#define BLKCOL 128                  // output cols per block
#define KC     32                   // k-chunk staged in LDS
#define LDA    36                   // lsA row stride (floats): 16B aligned, 64-bank conflict-free
#define LDB    136                  // lsB row stride (floats): 16B aligned, conflict-free
#define HROWS  (2 * NTOK + BLKTOK)  // compact h rows + tile padding

typedef float v2f __attribute__((ext_vector_type(2)));
typedef float v8f __attribute__((ext_vector_type(8)));

__device__ __forceinline__ v8f wmma_f32_k4(v2f a, v2f b, v8f c) {
  return __builtin_amdgcn_wmma_f32_16x16x4_f32(
      false, a, false, b, (short)0, c, false, false);
}

// ---------------------------------------------------------------------------
__global__ __launch_bounds__(256) void moe_init(int* __restrict__ counts,
                                                int* __restrict__ list_tok,
                                                float* __restrict__ list_p,
                                                float* __restrict__ out,
                                                size_t out_elems) {
  size_t i = (size_t)blockIdx.x * blockDim.x + threadIdx.x;
  size_t stride = (size_t)gridDim.x * blockDim.x;
  if (i < NEXP) counts[i] = 0;
  for (size_t j = i; j < (size_t)NEXP * SEGCAP; j += stride) {
    list_tok[j] = 0;
    list_p[j] = 0.0f;
  }
  for (size_t j = i; j < out_elems; j += stride) out[j] = 0.0f;
}

// ---------------------------------------------------------------------------
__global__ __launch_bounds__(256) void moe_router(const float* __restrict__ x,
                                                  const float* __restrict__ wr,
                                                  int* __restrict__ counts,
                                                  int* __restrict__ list_tok,
                                                  float* __restrict__ list_p) {
  int n = blockIdx.x * blockDim.x + threadIdx.x;
  if (n >= NTOK) return;
  const float* xr = x + (size_t)n * CDIM;

  float acc[NEXP];
#pragma unroll
  for (int e = 0; e < NEXP; ++e) acc[e] = 0.0f;

  for (int c = 0; c < CDIM; c += 4) {
    float4 xv = *(const float4*)(xr + c);
#pragma unroll
    for (int e = 0; e < NEXP; ++e) {
      float a = acc[e];
      a = fmaf(xv.x, wr[(size_t)(c + 0) * NEXP + e], a);
      a = fmaf(xv.y, wr[(size_t)(c + 1) * NEXP + e], a);
      a = fmaf(xv.z, wr[(size_t)(c + 2) * NEXP + e], a);
      a = fmaf(xv.w, wr[(size_t)(c + 3) * NEXP + e], a);
      acc[e] = a;
    }
  }

  // top-2, lowest index wins ties (matches jax.lax.top_k)
  int e0 = 0;
  float v0 = acc[0];
#pragma unroll
  for (int e = 1; e < NEXP; ++e)
    if (acc[e] > v0) { v0 = acc[e]; e0 = e; }
  int e1 = (e0 == 0) ? 1 : 0;
  float v1 = acc[e1];
#pragma unroll
  for (int e = 0; e < NEXP; ++e)
    if (e != e0 && acc[e] > v1) { v1 = acc[e]; e1 = e; }

  float ex = __expf(v1 - v0);
  float inv = 1.0f / (1.0f + ex);

  int pos0 = atomicAdd(&counts[e0], 1);
  list_tok[(size_t)e0 * SEGCAP + pos0] = n;
  list_p[(size_t)e0 * SEGCAP + pos0] = inv;
  int pos1 = atomicAdd(&counts[e1], 1);
  list_tok[(size_t)e1 * SEGCAP + pos1] = n;
  list_p[(size_t)e1 * SEGCAP + pos1] = ex * inv;
}

// ---------------------------------------------------------------------------
__global__ void moe_scan(const int* __restrict__ counts, int* __restrict__ offsets) {
  if (threadIdx.x == 0 && blockIdx.x == 0) {
    int s = 0;
    for (int e = 0; e < NEXP; ++e) {
      offsets[e] = s;
      s += counts[e];
    }
  }
}

// ---------------------------------------------------------------------------
// Shared GEMM body. 256 threads = 8 waves, block tile 128 tok x 128 col.
// Wave w: token subtile tm = w>>1 (32 rows), col subtile cn = w&1 (64 cols).
// Double-buffered LDS k-chunks of 32.
// ---------------------------------------------------------------------------
#define GEMM_BODY(EPILOGUE)                                                    \
  const int tid = threadIdx.x;                                                 \
  const int wave = tid >> 5, lane = tid & 31;                                  \
  const int half = lane >> 4, l15 = lane & 15;                                 \
  const int tm = wave >> 1, cn = wave & 1;                                     \
  /* staging coords */                                                         \
  const int arowi = tid >> 1, asub = tid & 1;       /* A: 2 thr/row, 16 f */   \
  const int bk = tid >> 3, bcol = (tid & 7) * 16;   /* B: 8 thr/row, 16 f */   \
  __shared__ float lsA[2][BLKTOK * LDA];                                       \
  __shared__ float lsB[2][KC * LDB];                                           \
  v8f acc[2][4];                                                               \
  _Pragma("unroll") for (int mt = 0; mt < 2; ++mt)                             \
  _Pragma("unroll") for (int nt = 0; nt < 4; ++nt) acc[mt][nt] = (v8f){};      \
  float4 sa[4], sb[4];                                                         \
  /* prologue: stage chunk 0 */                                                \
  _Pragma("unroll") for (int i = 0; i < 4; ++i) {                              \
    sa[i] = *(const float4*)(arow + asub * 16 + 4 * i);                        \
    sb[i] = *(const float4*)(bsrc + bcol + 4 * i);                             \
  }                                                                            \
  _Pragma("unroll") for (int i = 0; i < 4; ++i) {                              \
    *(float4*)(&lsA[0][arowi * LDA + asub * 16 + 4 * i]) = sa[i];              \
    *(float4*)(&lsB[0][bk * LDB + bcol + 4 * i]) = sb[i];                      \
  }                                                                            \
  __syncthreads();                                                             \
  const int nch = KDIM_ / KC;                                                  \
  for (int c = 0; c < nch; ++c) {                                              \
    const int cur = c & 1;                                                     \
    if (c + 1 < nch) { /* prefetch next chunk into regs */                     \
      const int kc = (c + 1) * KC;                                             \
      _Pragma("unroll") for (int i = 0; i < 4; ++i) {                          \
        sa[i] = *(const float4*)(arow + kc + asub * 16 + 4 * i);               \
        sb[i] = *(const float4*)(bsrc + (size_t)kc * LDBG + bcol + 4 * i);     \
      }                                                                        \
    }                                                                          \
    _Pragma("unroll") for (int kk = 0; kk < KC; kk += 4) {                     \
      const int ks = kk + (half << 1);                                         \
      v2f a0 = *(const v2f*)&lsA[cur][(tm * 32 + l15) * LDA + ks];             \
      v2f a1 = *(const v2f*)&lsA[cur][(tm * 32 + 16 + l15) * LDA + ks];        \
      _Pragma("unroll") for (int nt = 0; nt < 4; ++nt) {                       \
        const int col = cn * 64 + nt * 16 + l15;                               \
        v2f b;                                                                 \
        b.x = lsB[cur][ks * LDB + col];                                        \
        b.y = lsB[cur][(ks + 1) * LDB + col];                                  \
        acc[0][nt] = wmma_f32_k4(a0, b, acc[0][nt]);                           \
        acc[1][nt] = wmma_f32_k4(a1, b, acc[1][nt]);                           \
      }                                                                        \
    }                                                                          \
    if (c + 1 < nch) {                                                         \
      const int nxt = (c + 1) & 1;                                             \
      _Pragma("unroll") for (int i = 0; i < 4; ++i) {                          \
        *(float4*)(&lsA[nxt][arowi * LDA + asub * 16 + 4 * i]) = sa[i];        \
        *(float4*)(&lsB[nxt][bk * LDB + bcol + 4 * i]) = sb[i];                \
      }                                                                        \
    }                                                                          \
    __syncthreads();                                                           \
  }                                                                            \
  _Pragma("unroll") for (int mt = 0; mt < 2; ++mt)                             \
  _Pragma("unroll") for (int nt = 0; nt < 4; ++nt) {                           \
    const int col = colbase + cn * 64 + nt * 16 + l15;                         \
    _Pragma("unroll") for (int j = 0; j < 8; ++j) {                            \
      const int row = tok_base + tm * 32 + mt * 16 + j + half * 8;             \
      if (row < cnt) { EPILOGUE }                                              \
    }                                                                          \
  }

// GEMM1: h[row,:] = relu(x[tok(row),:] @ w_fc[e])^2
__global__ __launch_bounds__(256) void moe_gemm1(const float* __restrict__ x,
                                                 const float* __restrict__ w_fc,
                                                 const int* __restrict__ counts,
                                                 const int* __restrict__ offsets,
                                                 const int* __restrict__ list_tok,
                                                 float* __restrict__ h) {
  const int e = blockIdx.z;
  const int cnt = counts[e];
  const int tok_base = blockIdx.x * BLKTOK;
  if (tok_base >= cnt) return;  // block-uniform exit: EXEC all-ones for WMMA
  const int colbase = blockIdx.y * BLKCOL;
  const int ho = offsets[e];
  const int seg = e * SEGCAP;

  // per-thread A staging source (gathered token row) and B source
  const int atok = list_tok[seg + tok_base + (threadIdx.x >> 1)];
  const float* arow = x + (size_t)atok * CDIM;
  const float* bsrc = w_fc + (size_t)e * CDIM * DDIM +
                      (size_t)(threadIdx.x >> 3) * DDIM + colbase;
#define KDIM_ CDIM
#define LDBG  DDIM
  GEMM_BODY({
    float v = acc[mt][nt][j];
    v = fmaxf(v, 0.0f);
    v = v * v;
    h[(size_t)(ho + row) * DDIM + col] = v;
  })
#undef KDIM_
#undef LDBG
}

// GEMM2: out[tok(row),:] += p(row) * (h[row,:] @ w_proj[e])
__global__ __launch_bounds__(256) void moe_gemm2(const float* __restrict__ hbuf,
                                                 const float* __restrict__ w_proj,
                                                 const int* __restrict__ counts,
                                                 const int* __restrict__ offsets,
                                                 const int* __restrict__ list_tok,
                                                 const float* __restrict__ list_p,
                                                 float* __restrict__ out) {
  const int e = blockIdx.z;
  const int cnt = counts[e];
  const int tok_base = blockIdx.x * BLKTOK;
  if (tok_base >= cnt) return;
  const int colbase = blockIdx.y * BLKCOL;
  const int ho = offsets[e];
  const int seg = e * SEGCAP;

  const float* arow = hbuf + (size_t)(ho + tok_base + (threadIdx.x >> 1)) * DDIM;
  const float* bsrc = w_proj + (size_t)e * DDIM * CDIM +
                      (size_t)(threadIdx.x >> 3) * CDIM + colbase;
#define KDIM_ DDIM
#define LDBG  CDIM
  GEMM_BODY({
    const int tok = list_tok[seg + row];
    const float p = list_p[seg + row];
    atomicAdd(&out[(size_t)tok * CDIM + col], p * acc[mt][nt][j]);
  })
#undef KDIM_
#undef LDBG
}

// ---------------------------------------------------------------------------
// Host launcher. Workspace: 1 MiB header + (16384+128)*2048*4 B compact h.
// ---------------------------------------------------------------------------
extern "C" void kernel_launch(void* const* d_in, const int* in_sizes, int n_in,
                              void* d_out, int out_size, void* d_ws, size_t ws_size,
                              hipStream_t stream) {
  const float* x        = (const float*)d_in[0];
  const float* w_router = (const float*)d_in[1];
  const float* w_fc     = (const float*)d_in[2];
  const float* w_proj   = (const float*)d_in[3];
  float* out = (float*)d_out;

  char* ws = (char*)d_ws;
  int*   counts   = (int*)ws;
  int*   offsets  = counts + 8;
  int*   list_tok = (int*)(ws + 256);
  float* list_p   = (float*)(ws + 256 + NEXP * SEGCAP * 4);
  float* h        = (float*)(ws + (1u << 20));

  moe_init<<<2048, 256, 0, stream>>>(counts, list_tok, list_p, out, (size_t)out_size);
  moe_router<<<NTOK / 256, 256, 0, stream>>>(x, w_router, counts, list_tok, list_p);
  moe_scan<<<1, 32, 0, stream>>>(counts, offsets);

  dim3 g1(NTOK / BLKTOK, DDIM / BLKCOL, NEXP);
  moe_gemm1<<<g1, 256, 0, stream>>>(x, w_fc, counts, offsets, list_tok, h);
  dim3 g2(NTOK / BLKTOK, CDIM / BLKCOL, NEXP);
  moe_gemm2<<<g2, 256, 0, stream>>>(h, w_proj, counts, offsets, list_tok, list_p, out);
}